// PPNet_77833397338559
// MI455X (gfx1250) — compile-verified
//
#include <hip/hip_runtime.h>
#include <hip/hip_bf16.h>

typedef __attribute__((ext_vector_type(2))) float v2f;
typedef __attribute__((ext_vector_type(8))) float v8f;

#define HW_   196
#define NB_   64
// workspace float offsets (weights stored K-pair interleaved: wp[k/2][n][2])
#define OFF_W1T 0          // 256 pairs x 256 n x 2 = 131072
#define OFF_W2T 131072     // 128 pairs x 256 n x 2 =  65536
#define OFF_W3T 196608     // 128 pairs x 128 n x 2 =  32768
#define OFF_W4T 229376     //  64 pairs x 128 n x 2 =  16384
#define OFF_S1  245760     // [64][128]
#define OFF_SX2 253952     // [64][128]
#define WS_FLOATS 262144

static __device__ __forceinline__ v8f vzero8() {
    v8f z = {0.f,0.f,0.f,0.f,0.f,0.f,0.f,0.f};
    return z;
}

// ---------------------------------------------------------------------------
// Kernel 1: repack weights into K-pair-interleaved [K/2][N][2] layout so a
// B fragment (two consecutive K rows, same N) is one global_load_b64.
// Also zero the s1/sx2 accumulators.
// ---------------------------------------------------------------------------
__global__ void prep_kernel(const float* __restrict__ w1, const float* __restrict__ w2,
                            const float* __restrict__ w3, const float* __restrict__ w4,
                            float* __restrict__ ws) {
    int idx = blockIdx.x * 256 + threadIdx.x;
    if (idx < 196608) {                       // w1p: K=512,N=256 ; w2p: K=256,N=256
        int base = (idx < 131072) ? 0 : OFF_W2T;
        int e = idx - base;
        int j = e & 1, t = e >> 1;
        int n = t & 255, kp = t >> 8;
        const float* w = (idx < 131072) ? w1 : w2;
        int K = (idx < 131072) ? 512 : 256;
        ws[idx] = w[n * K + 2 * kp + j];
    } else if (idx < 229376) {                // w3p: K=256,N=128
        int e = idx - OFF_W3T;
        int j = e & 1, t = e >> 1;
        int n = t & 127, kp = t >> 7;
        ws[idx] = w3[n * 256 + 2 * kp + j];
    } else if (idx < 245760) {                // w4p: K=128,N=128
        int e = idx - OFF_W4T;
        int j = e & 1, t = e >> 1;
        int n = t & 127, kp = t >> 6;
        ws[idx] = w4[n * 128 + 2 * kp + j];
    } else if (idx < WS_FLOATS) {             // zero s1 / sx2 accumulators
        ws[idx] = 0.0f;
    }
}

// ---------------------------------------------------------------------------
// Kernel 2: fused 1x1-conv MLP chain with V_WMMA_F32_16X16X4_F32.
// Block = 256 threads (8 waves), handles 32 pixels (2 M-tiles) of one image.
// Grid = (7 chunks, 64 images). Produces s1/sx2 partial sums via atomicAdd.
// ---------------------------------------------------------------------------
__launch_bounds__(256)
__global__ void mlp_kernel(const float* __restrict__ fmap,
                           const float* __restrict__ bias1, const float* __restrict__ bias2,
                           const float* __restrict__ bias3, const float* __restrict__ bias4,
                           float* __restrict__ ws) {
    __shared__ float sX0[32][132];   // staged K-slice of input (128 ch), padded
    __shared__ float sX1[32][260];   // x1 (256 ch)
    __shared__ float sX2[32][260];   // x2 (256 ch)
    __shared__ float sX3[32][132];   // x3 (128 ch)
    __shared__ float sP1[256];
    __shared__ float sP2[256];

    const int ci   = blockIdx.x;          // pixel chunk 0..6 (32 pixels each)
    const int b    = blockIdx.y;          // image
    const int tid  = threadIdx.x;
    const int lane = tid & 31;
    const int w    = tid >> 5;            // wave id 0..7
    const int half = lane >> 4;           // 0: lanes 0-15, 1: lanes 16-31
    const int ml   = lane & 15;

    const v2f* w1p = (const v2f*)(ws + OFF_W1T);   // [kp][n] pairs
    const v2f* w2p = (const v2f*)(ws + OFF_W2T);
    const v2f* w3p = (const v2f*)(ws + OFF_W3T);
    const v2f* w4p = (const v2f*)(ws + OFF_W4T);

    int vr = HW_ - ci * 32; if (vr > 32) vr = 32;   // valid pixel rows in chunk

    // ---------------- GEMM1: x1 = relu(x0 * W1 + b1), K=512 in 4 slices of 128
    v8f a00 = vzero8(), a01 = vzero8();   // M-tile 0, N-tiles 0/1
    v8f a10 = vzero8(), a11 = vzero8();   // M-tile 1, N-tiles 0/1
    for (int ks = 0; ks < 4; ++ks) {
        __syncthreads();
        {   // stage fmap slice: sX0[p][c] = fmap[b][ks*128+c][ci*32+p] (0 for padded rows)
            int p = tid & 31, c0 = tid >> 5;     // c0 in 0..7
            for (int cc = c0; cc < 128; cc += 8) {
                float v = 0.f;
                if (p < vr)
                    v = fmap[((size_t)(b * 512 + ks * 128 + cc)) * HW_ + ci * 32 + p];
                sX0[p][cc] = v;
            }
        }
        __syncthreads();
        const int n0 = w * 32;
        #pragma unroll 8
        for (int kk = 0; kk < 32; ++kk) {
            const int k = kk * 4;
            v2f aL; aL.x = sX0[ml][k + 2 * half];      aL.y = sX0[ml][k + 2 * half + 1];
            v2f aH; aH.x = sX0[16 + ml][k + 2 * half]; aH.y = sX0[16 + ml][k + 2 * half + 1];
            const int kp = ks * 64 + kk * 2 + half;    // K-pair index
            v2f bA = w1p[kp * 256 + n0 + ml];
            v2f bB = w1p[kp * 256 + n0 + 16 + ml];
            a00 = __builtin_amdgcn_wmma_f32_16x16x4_f32(false, aL, false, bA, (short)0, a00, false, false);
            a01 = __builtin_amdgcn_wmma_f32_16x16x4_f32(false, aL, false, bB, (short)0, a01, false, false);
            a10 = __builtin_amdgcn_wmma_f32_16x16x4_f32(false, aH, false, bA, (short)0, a10, false, false);
            a11 = __builtin_amdgcn_wmma_f32_16x16x4_f32(false, aH, false, bB, (short)0, a11, false, false);
        }
    }
    __syncthreads();
    {   // bias + relu, store x1 to LDS (both M-tiles)
        const int n0 = w * 32;
        const float bvA = bias1[n0 + ml], bvB = bias1[n0 + 16 + ml];
        #pragma unroll
        for (int r = 0; r < 8; ++r) {
            const int m = r + 8 * half;
            sX1[m][n0 + ml]           = fmaxf(a00[r] + bvA, 0.f);
            sX1[m][n0 + 16 + ml]      = fmaxf(a01[r] + bvB, 0.f);
            sX1[16 + m][n0 + ml]      = fmaxf(a10[r] + bvA, 0.f);
            sX1[16 + m][n0 + 16 + ml] = fmaxf(a11[r] + bvB, 0.f);
        }
    }
    __syncthreads();

    // ---------------- GEMM2: x2 = relu(x1 * W2 + b2), K=256
    v8f c00 = vzero8(), c01 = vzero8(), c10 = vzero8(), c11 = vzero8();
    {
        const int n0 = w * 32;
        #pragma unroll 8
        for (int kk = 0; kk < 64; ++kk) {
            const int k = kk * 4;
            v2f aL; aL.x = sX1[ml][k + 2 * half];      aL.y = sX1[ml][k + 2 * half + 1];
            v2f aH; aH.x = sX1[16 + ml][k + 2 * half]; aH.y = sX1[16 + ml][k + 2 * half + 1];
            const int kp = kk * 2 + half;
            v2f bA = w2p[kp * 256 + n0 + ml];
            v2f bB = w2p[kp * 256 + n0 + 16 + ml];
            c00 = __builtin_amdgcn_wmma_f32_16x16x4_f32(false, aL, false, bA, (short)0, c00, false, false);
            c01 = __builtin_amdgcn_wmma_f32_16x16x4_f32(false, aL, false, bB, (short)0, c01, false, false);
            c10 = __builtin_amdgcn_wmma_f32_16x16x4_f32(false, aH, false, bA, (short)0, c10, false, false);
            c11 = __builtin_amdgcn_wmma_f32_16x16x4_f32(false, aH, false, bB, (short)0, c11, false, false);
        }
        const float bvA = bias2[n0 + ml], bvB = bias2[n0 + 16 + ml];
        __syncthreads();
        #pragma unroll
        for (int r = 0; r < 8; ++r) {
            const int m = r + 8 * half;
            sX2[m][n0 + ml]           = fmaxf(c00[r] + bvA, 0.f);
            sX2[m][n0 + 16 + ml]      = fmaxf(c01[r] + bvB, 0.f);
            sX2[16 + m][n0 + ml]      = fmaxf(c10[r] + bvA, 0.f);
            sX2[16 + m][n0 + 16 + ml] = fmaxf(c11[r] + bvB, 0.f);
        }
    }
    __syncthreads();

    // ---------------- GEMM3: x3 = relu(x2 * W3 + b3), K=256, N=128 (1 N-tile/wave)
    v8f d0 = vzero8(), d1 = vzero8();
    {
        const int n0 = w * 16;
        #pragma unroll 8
        for (int kk = 0; kk < 64; ++kk) {
            const int k = kk * 4;
            v2f aL; aL.x = sX2[ml][k + 2 * half];      aL.y = sX2[ml][k + 2 * half + 1];
            v2f aH; aH.x = sX2[16 + ml][k + 2 * half]; aH.y = sX2[16 + ml][k + 2 * half + 1];
            const int kp = kk * 2 + half;
            v2f bA = w3p[kp * 128 + n0 + ml];
            d0 = __builtin_amdgcn_wmma_f32_16x16x4_f32(false, aL, false, bA, (short)0, d0, false, false);
            d1 = __builtin_amdgcn_wmma_f32_16x16x4_f32(false, aH, false, bA, (short)0, d1, false, false);
        }
        const float bv = bias3[n0 + ml];
        __syncthreads();
        #pragma unroll
        for (int r = 0; r < 8; ++r) {
            const int m = r + 8 * half;
            sX3[m][n0 + ml]      = fmaxf(d0[r] + bv, 0.f);
            sX3[16 + m][n0 + ml] = fmaxf(d1[r] + bv, 0.f);
        }
    }
    __syncthreads();

    // ---------------- GEMM4: x4 = sigmoid(x3 * W4 + b4), K=128, N=128
    v8f e0 = vzero8(), e1 = vzero8();
    {
        const int n0 = w * 16;
        #pragma unroll 8
        for (int kk = 0; kk < 32; ++kk) {
            const int k = kk * 4;
            v2f aL; aL.x = sX3[ml][k + 2 * half];      aL.y = sX3[ml][k + 2 * half + 1];
            v2f aH; aH.x = sX3[16 + ml][k + 2 * half]; aH.y = sX3[16 + ml][k + 2 * half + 1];
            const int kp = kk * 2 + half;
            v2f bA = w4p[kp * 128 + n0 + ml];
            e0 = __builtin_amdgcn_wmma_f32_16x16x4_f32(false, aL, false, bA, (short)0, e0, false, false);
            e1 = __builtin_amdgcn_wmma_f32_16x16x4_f32(false, aH, false, bA, (short)0, e1, false, false);
        }
        // sigmoid + masked reduction over valid pixel rows (both M-tiles)
        const float bv = bias4[n0 + ml];
        float s1p = 0.f, s2p = 0.f;
        #pragma unroll
        for (int r = 0; r < 8; ++r) {
            const int m = r + 8 * half;
            float v0 = 1.f / (1.f + __expf(-(e0[r] + bv)));
            float v1 = 1.f / (1.f + __expf(-(e1[r] + bv)));
            if (m < vr)      { s1p += v0; s2p += v0 * v0; }
            if (16 + m < vr) { s1p += v1; s2p += v1 * v1; }
        }
        sP1[half * 128 + n0 + ml] = s1p;
        sP2[half * 128 + n0 + ml] = s2p;
    }
    __syncthreads();
    if (tid < 128) {
        float s1 = sP1[tid] + sP1[128 + tid];
        float s2 = sP2[tid] + sP2[128 + tid];
        atomicAdd(&ws[OFF_S1  + b * 128 + tid], s1);
        atomicAdd(&ws[OFF_SX2 + b * 128 + tid], s2);
    }
}

// ---------------------------------------------------------------------------
// Kernel 3: prototype distances, min over channels, log-act, class max, logits
// One block per image.
// ---------------------------------------------------------------------------
__launch_bounds__(256)
__global__ void head_kernel(const float* __restrict__ protos, const float* __restrict__ last_w,
                            const float* __restrict__ ws, float* __restrict__ out) {
    __shared__ float ls1[128], ls2[128], lmax[200];
    const int b = blockIdx.x, tid = threadIdx.x;
    if (tid < 128) {
        ls1[tid] = ws[OFF_S1  + b * 128 + tid];
        ls2[tid] = ws[OFF_SX2 + b * 128 + tid];
    }
    __syncthreads();
    float* min_out = out + NB_ * 200;            // min_d region after logits
    if (tid < 200) {
        float mx = -3.4e38f;
        for (int pp = 0; pp < 10; ++pp) {
            const int p = tid * 10 + pp;
            const float* pr = protos + (size_t)p * 128;
            float mind = 3.4e38f;
            #pragma unroll 4
            for (int c = 0; c < 128; ++c) {
                const float q = pr[c];
                float dd = ls2[c] - 2.f * ls1[c] * q + 196.f * q * q;
                dd = fmaxf(dd, 0.f);
                mind = fminf(mind, dd);
            }
            min_out[b * 2000 + p] = mind;
            const float act = logf((mind + 1.f) / (mind + 1e-4f));
            mx = fmaxf(mx, act);
        }
        lmax[tid] = mx;
    }
    __syncthreads();
    if (tid < 200) {
        float acc = 0.f;
        for (int j = 0; j < 200; ++j) acc += lmax[j] * last_w[tid * 200 + j];
        out[b * 200 + tid] = acc;
    }
}

// ---------------------------------------------------------------------------
extern "C" void kernel_launch(void* const* d_in, const int* in_sizes, int n_in,
                              void* d_out, int out_size, void* d_ws, size_t ws_size,
                              hipStream_t stream) {
    const float* fmap   = (const float*)d_in[0];
    const float* w1     = (const float*)d_in[1];
    const float* b1     = (const float*)d_in[2];
    const float* w2     = (const float*)d_in[3];
    const float* b2     = (const float*)d_in[4];
    const float* w3     = (const float*)d_in[5];
    const float* b3     = (const float*)d_in[6];
    const float* w4     = (const float*)d_in[7];
    const float* b4     = (const float*)d_in[8];
    const float* protos = (const float*)d_in[9];
    const float* last_w = (const float*)d_in[10];
    float* ws  = (float*)d_ws;
    float* out = (float*)d_out;

    prep_kernel<<<WS_FLOATS / 256, 256, 0, stream>>>(w1, w2, w3, w4, ws);
    mlp_kernel<<<dim3(7, 64), 256, 0, stream>>>(fmap, b1, b2, b3, b4, ws);
    head_kernel<<<NB_, 256, 0, stream>>>(protos, last_w, ws, out);
}